// LPSolverADMM_58961311039937
// MI455X (gfx1250) — compile-verified
//
#include <hip/hip_runtime.h>
#include <hip/hip_bf16.h>

// ---------------------------------------------------------------------------
// ADMM LP solver on gfx1250 (MI455X).
// A: 2048x4096 f32.  30 ADMM iters x 15 CG iters.  All GEMVs via
// V_WMMA_F32_16X16X4_F32 (16 rows per block, K-split x8 across waves, LDS
// combine).  A^T is materialized once per call so both matvec directions are
// coalesced.  A + A^T = 67 MB < 192 MB L2: the whole solve is L2-resident.
// Hot loop: 8 b64 A-loads + 8 b64 vec-loads in flight, then 8 WMMAs on two
// independent accumulator tiles (uniform compile-time trip count).
// ---------------------------------------------------------------------------

#define LP_M 2048      // rows of A
#define LP_N 4096      // cols of A
#define LP_ADMM 30
#define LP_CG 15
#define LP_NSCAL (LP_ADMM * 31)   // per step: rz[0..15] (16) + pKp[0..14] (15)

typedef __attribute__((ext_vector_type(2))) float v2f;
typedef __attribute__((ext_vector_type(8))) float v8f;

// ---------------------------------------------------------------------------
// WMMA GEMV core: one 256-thread block handles 16 rows of `mat` (row-major,
// COLS columns). Wave w of 8 handles K-segment w.  B's 16 columns all carry
// the same broadcast vector chunk, so every column of D holds the row dots.
//
// A-matrix VGPR layout (16x4 f32): lane L -> (M = L&15, K = 2*(L>>4)+{0,1}).
// B-matrix (4x16 f32, broadcast cols): lane L -> v[K], same K index as A.
// D (16x16 f32): lane 0 holds M=0..7 in acc[0..7], lane 16 holds M=8..15.
// ---------------------------------------------------------------------------
template <int COLS>
__device__ __forceinline__ void gemv_core_store(
    const float* __restrict__ mat, const float* __restrict__ vec,
    int rowbase, float* __restrict__ red /* [8][16] LDS */) {
  constexpr int SEG = COLS / 8;   // columns per wave (multiple of 32)
  constexpr int U   = 8;          // WMMAs per unrolled iteration (32 cols)

  const int tid  = threadIdx.x;
  const int w    = tid >> 5;      // wave in block: K segment
  const int lane = tid & 31;
  const int half = lane >> 4;
  const int mrow = lane & 15;

  const float* mp = mat + (size_t)(rowbase + mrow) * (size_t)COLS
                        + (size_t)w * SEG + (half << 1);
  const float* vp = vec + (size_t)w * SEG + (half << 1);

  v8f acc0{}, acc1{};
  for (int k = 0; k < SEG; k += 4 * U) {   // uniform scalar trip count
    v2f a[U], b[U];
#pragma unroll
    for (int u = 0; u < U; ++u) {
      a[u] = *(const v2f*)(mp + k + 4 * u);
      b[u] = *(const v2f*)(vp + k + 4 * u);
    }
#pragma unroll
    for (int u = 0; u < U; ++u) {
      if (u & 1)
        acc1 = __builtin_amdgcn_wmma_f32_16x16x4_f32(
            false, a[u], false, b[u], (short)0, acc1, false, false);
      else
        acc0 = __builtin_amdgcn_wmma_f32_16x16x4_f32(
            false, a[u], false, b[u], (short)0, acc0, false, false);
    }
  }
  if (mrow == 0) {
    // lane 0: rows 0..7, lane 16: rows 8..15
#pragma unroll
    for (int r = 0; r < 8; ++r) red[w * 16 + half * 8 + r] = acc0[r] + acc1[r];
  }
}

// MODE 0: out[row] = res
// MODE 1: kp = rho*res + sig*aux1[row]; out[row]=kp; dot += aux1[row]*kp
// MODE 2: out[row] = sig*aux1[row] - aux2[row] + res          (right vector)
// MODE 3: g = rho*res + sig*aux2[row]; r0 = aux1[row]-g; out[row]=r0;
//         z0 = r0/aux3[row]; out2[row]=z0; dot += r0*z0       (CG init)
template <int MODE, int COLS>
__global__ void __launch_bounds__(256)
gemv16(const float* __restrict__ mat, const float* __restrict__ vec,
       float* __restrict__ out,
       const float* __restrict__ aux1, const float* __restrict__ aux2,
       const float* __restrict__ aux3, float* __restrict__ out2,
       float* __restrict__ dot_out,
       const float* __restrict__ rho_p, const float* __restrict__ sig_p) {
  __shared__ float red[8 * 16];
  const int rowbase = blockIdx.x * 16;
  gemv_core_store<COLS>(mat, vec, rowbase, red);
  __syncthreads();

  const int tid = threadIdx.x;
  float local = 0.0f;
  if (tid < 16) {
    float res = 0.0f;
#pragma unroll
    for (int w = 0; w < 8; ++w) res += red[w * 16 + tid];
    const int row = rowbase + tid;
    if (MODE == 0) {
      out[row] = res;
    } else if (MODE == 1) {
      const float rho = rho_p[0], sig = sig_p[0];
      const float pv = aux1[row];
      const float kp = rho * res + sig * pv;
      out[row] = kp;
      local = pv * kp;
    } else if (MODE == 2) {
      const float sig = sig_p[0];
      out[row] = sig * aux1[row] - aux2[row] + res;
    } else if (MODE == 3) {
      const float rho = rho_p[0], sig = sig_p[0];
      const float g  = rho * res + sig * aux2[row];
      const float r0 = aux1[row] - g;
      out[row] = r0;
      const float z0 = r0 / aux3[row];
      out2[row] = z0;
      local = r0 * z0;
    }
  }
  if (MODE == 1 || MODE == 3) {
    if (tid < 32) {  // whole wave 0 participates in the shuffle reduction
      float v = local;
#pragma unroll
      for (int off = 16; off > 0; off >>= 1) v += __shfl_down(v, off);
      if (tid == 0) atomicAdd(dot_out, v);
    }
  }
}

// ---------------------------------------------------------------------------
// Tiled transpose: AT[j*M + i] = A[i*N + j].  block (32,8), grid (N/32, M/32).
// ---------------------------------------------------------------------------
__global__ void __launch_bounds__(256)
k_transpose(const float* __restrict__ A, float* __restrict__ AT) {
  __shared__ float tile[32][33];
  const int tx = threadIdx.x, ty = threadIdx.y;
  const int x0 = blockIdx.x * 32;   // col of A
  const int y0 = blockIdx.y * 32;   // row of A
#pragma unroll
  for (int k = 0; k < 32; k += 8)
    tile[ty + k][tx] = A[(size_t)(y0 + ty + k) * LP_N + (x0 + tx)];
  __syncthreads();
#pragma unroll
  for (int k = 0; k < 32; k += 8)
    AT[(size_t)(x0 + ty + k) * LP_M + (y0 + tx)] = tile[tx][ty + k];
}

// Mdiag[j] = sig + rho * sum_i AT[j][i]^2   (one wave per AT row, coalesced)
__global__ void __launch_bounds__(256)
k_mdiag(const float* __restrict__ AT, float* __restrict__ Mdiag,
        const float* __restrict__ rho_p, const float* __restrict__ sig_p) {
  const int gtid = blockIdx.x * blockDim.x + threadIdx.x;
  const int j    = gtid >> 5;
  const int lane = gtid & 31;
  if (j >= LP_N) return;
  const float* row = AT + (size_t)j * LP_M;
  float s = 0.0f;
  for (int i = lane; i < LP_M; i += 32) { float a = row[i]; s += a * a; }
#pragma unroll
  for (int off = 16; off > 0; off >>= 1) s += __shfl_down(s, off);
  if (lane == 0) Mdiag[j] = sig_p[0] + rho_p[0] * s;
}

// Scaled problem data + zero state + zero scalar accumulators.
__global__ void __launch_bounds__(256)
k_setup(const float* __restrict__ c, const float* __restrict__ lb,
        const float* __restrict__ ub, const float* __restrict__ d,
        const float* __restrict__ e,
        const float* __restrict__ gc_p, const float* __restrict__ gb_p,
        float* __restrict__ c_s, float* __restrict__ lb_s,
        float* __restrict__ ub_s, float* __restrict__ x,
        float* __restrict__ z, float* __restrict__ y,
        float* __restrict__ xt, float* __restrict__ sc) {
  const int j = blockIdx.x * blockDim.x + threadIdx.x;
  if (j < LP_N) {
    c_s[j] = gc_p[0] * d[j] * c[j];
    x[j] = 0.0f; xt[j] = 0.0f;
  }
  if (j < LP_M) {
    const float gb = gb_p[0];
    lb_s[j] = gb * e[j] * lb[j];
    ub_s[j] = gb * e[j] * ub[j];
    z[j] = 0.0f; y[j] = 0.0f;
  }
  if (j < LP_NSCAL) sc[j] = 0.0f;
}

// q = rho*z - y
__global__ void __launch_bounds__(256)
k_q(const float* __restrict__ z, const float* __restrict__ y,
    const float* __restrict__ rho_p, float* __restrict__ q) {
  const int i = blockIdx.x * blockDim.x + threadIdx.x;
  if (i < LP_M) q[i] = rho_p[0] * z[i] - y[i];
}

// CG inner update: a = rz_old/pKp; xt+=a*p; r-=a*Kp; zv=r/Mdiag; rz_new+=r*zv
__global__ void __launch_bounds__(256)
k_cg_update(float* __restrict__ xt, float* __restrict__ r,
            const float* __restrict__ p, const float* __restrict__ Kp,
            const float* __restrict__ Mdiag,
            const float* __restrict__ rz_old_p, const float* __restrict__ pkp_p,
            float* __restrict__ rz_new_p, float* __restrict__ zv) {
  __shared__ float sred[8];
  const int j = blockIdx.x * blockDim.x + threadIdx.x;
  const float a = rz_old_p[0] / pkp_p[0];
  float local = 0.0f;
  if (j < LP_N) {
    xt[j] += a * p[j];
    const float rn = r[j] - a * Kp[j];
    r[j] = rn;
    const float zn = rn / Mdiag[j];
    zv[j] = zn;
    local = rn * zn;
  }
  float v = local;
#pragma unroll
  for (int off = 16; off > 0; off >>= 1) v += __shfl_down(v, off);
  if ((threadIdx.x & 31) == 0) sred[threadIdx.x >> 5] = v;
  __syncthreads();
  if (threadIdx.x == 0) {
    float s = 0.0f;
#pragma unroll
    for (int w = 0; w < 8; ++w) s += sred[w];
    atomicAdd(rz_new_p, s);
  }
}

// p = zv + (rz_new/rz_old) * p
__global__ void __launch_bounds__(256)
k_p_update(float* __restrict__ p, const float* __restrict__ zv,
           const float* __restrict__ rz_new_p, const float* __restrict__ rz_old_p) {
  const int j = blockIdx.x * blockDim.x + threadIdx.x;
  if (j < LP_N) {
    const float beta = rz_new_p[0] / rz_old_p[0];
    p[j] = zv[j] + beta * p[j];
  }
}

// x = al*xt + (1-al)*x
__global__ void __launch_bounds__(256)
k_x_relax(float* __restrict__ x, const float* __restrict__ xt,
          const float* __restrict__ al_p) {
  const int j = blockIdx.x * blockDim.x + threadIdx.x;
  if (j < LP_N) {
    const float al = al_p[0];
    x[j] = al * xt[j] + (1.0f - al) * x[j];
  }
}

// ztr = al*zt_raw + (1-al)*z;  z = clip(ztr + y/rho, lb_s, ub_s);
// y += rho*(ztr - z)
__global__ void __launch_bounds__(256)
k_zy(float* __restrict__ z, float* __restrict__ y,
     const float* __restrict__ zt_raw, const float* __restrict__ lb_s,
     const float* __restrict__ ub_s, const float* __restrict__ rho_p,
     const float* __restrict__ al_p) {
  const int i = blockIdx.x * blockDim.x + threadIdx.x;
  if (i < LP_M) {
    const float rho = rho_p[0], al = al_p[0];
    const float ztr = al * zt_raw[i] + (1.0f - al) * z[i];
    float zn = ztr + y[i] / rho;
    zn = fminf(fmaxf(zn, lb_s[i]), ub_s[i]);
    y[i] += rho * (ztr - zn);
    z[i] = zn;
  }
}

// out = x * d / gb
__global__ void __launch_bounds__(256)
k_out(const float* __restrict__ x, const float* __restrict__ d,
      const float* __restrict__ gb_p, float* __restrict__ out) {
  const int j = blockIdx.x * blockDim.x + threadIdx.x;
  if (j < LP_N) out[j] = x[j] * d[j] / gb_p[0];
}

// ---------------------------------------------------------------------------
// Host orchestration (graph-capturable: only stream launches).
// ---------------------------------------------------------------------------
extern "C" void kernel_launch(void* const* d_in, const int* in_sizes, int n_in,
                              void* d_out, int out_size, void* d_ws, size_t ws_size,
                              hipStream_t stream) {
  const float* A   = (const float*)d_in[0];
  const float* c   = (const float*)d_in[1];
  const float* lb  = (const float*)d_in[2];
  const float* ub  = (const float*)d_in[3];
  const float* dd  = (const float*)d_in[4];
  const float* e   = (const float*)d_in[5];
  const float* gc  = (const float*)d_in[6];
  const float* gb  = (const float*)d_in[7];
  const float* rho = (const float*)d_in[8];
  const float* sig = (const float*)d_in[9];
  const float* al  = (const float*)d_in[10];
  float* out = (float*)d_out;
  float* ws  = (float*)d_ws;

  // workspace layout (floats)
  size_t off = 0;
  float* AT    = ws + off; off += (size_t)LP_N * LP_M;
  float* Mdiag = ws + off; off += LP_N;
  float* c_s   = ws + off; off += LP_N;
  float* lb_s  = ws + off; off += LP_M;
  float* ub_s  = ws + off; off += LP_M;
  float* x     = ws + off; off += LP_N;
  float* z     = ws + off; off += LP_M;
  float* y     = ws + off; off += LP_M;
  float* xt    = ws + off; off += LP_N;
  float* r     = ws + off; off += LP_N;
  float* p     = ws + off; off += LP_N;
  float* zv    = ws + off; off += LP_N;
  float* t     = ws + off; off += LP_M;   // A @ v results
  float* q     = ws + off; off += LP_M;
  float* Kp    = ws + off; off += LP_N;
  float* right = ws + off; off += LP_N;
  float* sc    = ws + off; off += LP_NSCAL;  // scalar accumulators

  const dim3 B256(256);
  const dim3 GN(LP_N / 256);          // elementwise over n
  const dim3 GM(LP_M / 256);          // elementwise over m
  const dim3 G_A(LP_M / 16);          // gemv over A   (rows=M, cols=N)
  const dim3 G_AT(LP_N / 16);         // gemv over A^T (rows=N, cols=M)

  // ---- one-time-per-call setup ----
  hipLaunchKernelGGL(k_transpose, dim3(LP_N / 32, LP_M / 32), dim3(32, 8), 0,
                     stream, A, AT);
  hipLaunchKernelGGL(k_mdiag, dim3((LP_N * 32) / 256), B256, 0, stream,
                     AT, Mdiag, rho, sig);
  hipLaunchKernelGGL(k_setup, GN, B256, 0, stream, c, lb, ub, dd, e, gc, gb,
                     c_s, lb_s, ub_s, x, z, y, xt, sc);

  // ---- ADMM loop ----
  for (int s = 0; s < LP_ADMM; ++s) {
    float* rz_base  = sc + (size_t)s * 31;       // rz[0..15]
    float* pkp_base = rz_base + 16;              // pKp[0..14]

    // right = sig*x - c_s + A^T (rho*z - y)
    hipLaunchKernelGGL(k_q, GM, B256, 0, stream, z, y, rho, q);
    hipLaunchKernelGGL((gemv16<2, LP_M>), G_AT, B256, 0, stream,
                       AT, q, right, x, c_s, nullptr, nullptr, nullptr,
                       rho, sig);

    // CG init: r = right - (rho*A^T(A xt) + sig*xt); p = zv = r/Mdiag; rz0
    hipLaunchKernelGGL((gemv16<0, LP_N>), G_A, B256, 0, stream,
                       A, xt, t, nullptr, nullptr, nullptr, nullptr,
                       nullptr, rho, sig);
    hipLaunchKernelGGL((gemv16<3, LP_M>), G_AT, B256, 0, stream,
                       AT, t, r, right, xt, Mdiag, p, rz_base,
                       rho, sig);

    // CG iterations
    for (int i = 0; i < LP_CG; ++i) {
      hipLaunchKernelGGL((gemv16<0, LP_N>), G_A, B256, 0, stream,
                         A, p, t, nullptr, nullptr, nullptr, nullptr,
                         nullptr, rho, sig);
      hipLaunchKernelGGL((gemv16<1, LP_M>), G_AT, B256, 0, stream,
                         AT, t, Kp, p, nullptr, nullptr, nullptr,
                         pkp_base + i, rho, sig);
      hipLaunchKernelGGL(k_cg_update, GN, B256, 0, stream,
                         xt, r, p, Kp, Mdiag, rz_base + i, pkp_base + i,
                         rz_base + i + 1, zv);
      hipLaunchKernelGGL(k_p_update, GN, B256, 0, stream,
                         p, zv, rz_base + i + 1, rz_base + i);
    }

    // ztilde_raw = A xt ; relax + clip + dual update
    hipLaunchKernelGGL((gemv16<0, LP_N>), G_A, B256, 0, stream,
                       A, xt, t, nullptr, nullptr, nullptr, nullptr,
                       nullptr, rho, sig);
    hipLaunchKernelGGL(k_x_relax, GN, B256, 0, stream, x, xt, al);
    hipLaunchKernelGGL(k_zy, GM, B256, 0, stream, z, y, t, lb_s, ub_s, rho, al);
  }

  hipLaunchKernelGGL(k_out, GN, B256, 0, stream, x, dd, gb, out);

  (void)in_sizes; (void)n_in; (void)out_size; (void)ws_size;
}